// MedianPool3d_68994354642979
// MI455X (gfx1250) — compile-verified
//
#include <hip/hip_runtime.h>
#include <cstdint>
#include <cstddef>

#define N3D 160
#define TS 16
#define LT 18           // 16 + 2 halo
#define ZCHUNK 32       // 160 / 32 = 5 z-chunks

// ---- gfx1250 async global->LDS path (probe-guarded, falls back to plain loads)
#ifndef MED3D_ASYNC
#if defined(__has_builtin)
#if __has_builtin(__builtin_amdgcn_global_load_async_to_lds_b32) && \
    __has_builtin(__builtin_amdgcn_s_wait_asynccnt)
#define MED3D_ASYNC 1
#endif
#endif
#endif
#ifndef MED3D_ASYNC
#define MED3D_ASYNC 0
#endif

#define AS1 __attribute__((address_space(1)))
#define AS3 __attribute__((address_space(3)))

__device__ __forceinline__ int refl(int i) {
  i = (i < 0) ? -i : i;
  return (i > N3D - 1) ? (2 * (N3D - 1) - i) : i;
}

// compare-exchange: a <- min, b <- max  (v_min_num_f32 / v_max_num_f32)
__device__ __forceinline__ void ce(float& a, float& b) {
  float lo = fminf(a, b);
  float hi = fmaxf(a, b);
  a = lo; b = hi;
}

__device__ __forceinline__ void sort3(float& a, float& b, float& c) {
  ce(a, b); ce(b, c); ce(a, b);
}

// ---- Batcher odd-even merges (all inputs sorted ascending) ----

// merge sorted (x0,x1) with (y0) -> f[0..2]   (2 CE)
__device__ __forceinline__ void merge21(float x0, float x1, float y0, float f[3]) {
  float h0 = x0, h1 = y0; ce(h0, h1);
  f[0] = h0; f[1] = x1; f[2] = h1; ce(f[1], f[2]);
}

// merge sorted (x0..x2) with (y0) -> f[0..3]  (3 CE)
__device__ __forceinline__ void merge31(float x0, float x1, float x2, float y0, float f[4]) {
  float h[3]; merge21(x0, x2, y0, h);
  f[0] = h[0]; f[1] = x1; f[2] = h[1]; ce(f[1], f[2]); f[3] = h[2];
}

// merge sorted (x0..x2) with (y0,y1) -> e[0..4]  (5 CE)
__device__ __forceinline__ void merge32(float x0, float x1, float x2, float y0, float y1, float e[5]) {
  float fE[3]; merge21(x0, x2, y0, fE);
  float g0 = x1, g1 = y1; ce(g0, g1);
  e[0] = fE[0];
  e[1] = g0; e[2] = fE[1]; ce(e[1], e[2]);
  e[3] = g1; e[4] = fE[2]; ce(e[3], e[4]);
}

// merge sorted (u0,u1) with (v0,v1) -> g[0..3]   (3 CE)
__device__ __forceinline__ void merge22(float u0, float u1, float v0, float v1, float g[4]) {
  float e0 = u0, e1 = v0; ce(e0, e1);
  float o0 = u1, o1 = v1; ce(o0, o1);
  g[0] = e0; g[1] = o0; g[2] = e1; ce(g[1], g[2]); g[3] = o1;
}

// merge sorted (u0..u2) with (v0..v2) -> p[0..5]  (6 CE)
__device__ __forceinline__ void merge33(float u0, float u1, float u2,
                                        float v0, float v1, float v2, float p[6]) {
  float g[4]; merge22(u0, u2, v0, v2, g);
  float h0 = u1, h1 = v1; ce(h0, h1);
  p[0] = g[0];
  p[1] = h0; p[2] = g[1]; ce(p[1], p[2]);
  p[3] = h1; p[4] = g[2]; ce(p[3], p[4]);
  p[5] = g[3];
}

// merge sorted a[0..5] with (b0..b2) -> r[0..8]   (12 CE)
__device__ __forceinline__ void merge63(const float a[6], float b0, float b1, float b2, float r[9]) {
  float E[5]; merge32(a[0], a[2], a[4], b0, b2, E);
  float O[4]; merge31(a[1], a[3], a[5], b1, O);
  r[0] = E[0];
  r[1] = O[0]; r[2] = E[1]; ce(r[1], r[2]);
  r[3] = O[1]; r[4] = E[2]; ce(r[3], r[4]);
  r[5] = O[2]; r[6] = E[3]; ce(r[5], r[6]);
  r[7] = O[3]; r[8] = E[4]; ce(r[7], r[8]);
}

// Pruned Batcher merge(9,9): produce only ranks 4..13 of the sorted union
// into m[0..9]  (48 min/max ops vs 60 for the full merge).
// Derived by dead-output elimination from the full odd-even network;
// verified: a=odds(1..17), b=evens(2..18) -> m = 5..14.
__device__ __forceinline__ void merge99_mid(const float a[9], const float b[9], float m[10]) {
  // ---- E = merge55 of evens (a0,a2,a4,a6,a8 | b0,b2,b4,b6,b8), need E[2..7]
  //   inner merge33(a0,a4,a8 | b0,b4,b8), need p[1..4]:
  float e1 = fmaxf(a[0], b[0]);
  float o0 = fminf(a[8], b[8]);
  float g1 = o0, g2 = e1; ce(g1, g2);
  float h0 = a[4], h1 = b[4]; ce(h0, h1);
  float p1 = h0, p2 = g1; ce(p1, p2);
  float p3 = h1, p4 = g2; ce(p3, p4);
  //   inner merge22(a2,a6 | b2,b6), full q[0..3]:
  float q[4]; merge22(a[2], a[6], b[2], b[6], q);
  //   combine -> E2..E7:
  float E2 = fmaxf(q[0], p1);
  float E3 = q[1], E4 = p2; ce(E3, E4);
  float E5 = q[2], E6 = p3; ce(E5, E6);
  float E7 = fminf(q[3], p4);
  // ---- O = merge44 of odds (a1,a3,a5,a7 | b1,b3,b5,b7), need O[1..6]
  //   E' = merge22(a1,a5 | b1,b5), need E'1..E'3:
  float pe1 = fmaxf(a[1], b[1]);
  float po0 = a[5], po1 = b[5]; ce(po0, po1);
  float Ep1 = po0, Ep2 = pe1; ce(Ep1, Ep2);
  float Ep3 = po1;
  //   O' = merge22(a3,a7 | b3,b7), need O'0..O'2:
  float qe0 = a[3], qe1 = b[3]; ce(qe0, qe1);
  float qo0 = fminf(a[7], b[7]);
  float Op0 = qe0;
  float Op1 = qo0, Op2 = qe1; ce(Op1, Op2);
  //   combine -> O1..O6:
  float O1 = Op0, O2 = Ep1; ce(O1, O2);
  float O3 = Op1, O4 = Ep2; ce(O3, O4);
  float O5 = Op2, O6 = Ep3; ce(O5, O6);
  // ---- final layer: m[0..9] = ranks 4..13
  m[0] = fmaxf(O1, E2);                  // r4
  m[1] = O2; m[2] = E3; ce(m[1], m[2]);  // r5, r6
  m[3] = O3; m[4] = E4; ce(m[3], m[4]);  // r7, r8
  m[5] = O4; m[6] = E5; ce(m[5], m[6]);  // r9, r10
  m[7] = O5; m[8] = E6; ce(m[7], m[8]);  // r11, r12
  m[9] = fminf(O6, E7);                  // r13
}

// read 3x3 neighborhood from LDS plane tile and produce sorted 9-list (27 CE)
__device__ __forceinline__ void sort9_from(const float* bb, int ty, int tx, float s[9]) {
  float v0 = bb[(ty    ) * LT + tx    ];
  float v1 = bb[(ty    ) * LT + tx + 1];
  float v2 = bb[(ty    ) * LT + tx + 2];
  float v3 = bb[(ty + 1) * LT + tx    ];
  float v4 = bb[(ty + 1) * LT + tx + 1];
  float v5 = bb[(ty + 1) * LT + tx + 2];
  float v6 = bb[(ty + 2) * LT + tx    ];
  float v7 = bb[(ty + 2) * LT + tx + 1];
  float v8 = bb[(ty + 2) * LT + tx + 2];
  sort3(v0, v1, v2);
  sort3(v3, v4, v5);
  sort3(v6, v7, v8);
  float s6[6];
  merge33(v0, v1, v2, v3, v4, v5, s6);
  merge63(s6, v6, v7, v8, s);
}

extern "C" __global__ void __launch_bounds__(TS * TS)
median_pool3d_68994354642979_kernel(const float* __restrict__ x, float* __restrict__ out) {
  __shared__ float buf[2][LT * LT];

  const int tx  = threadIdx.x;
  const int ty  = threadIdx.y;
  const int tid = ty * TS + tx;
  const int ntx = N3D / TS;                   // 10 tiles per row
  const int gx0 = (blockIdx.x % ntx) * TS;
  const int gy0 = (blockIdx.x / ntx) * TS;
  const int z0  = blockIdx.y * ZCHUNK;
  const int b   = blockIdx.z;

  const size_t plane = (size_t)N3D * N3D;
  const float* xb = x   + (size_t)b * N3D * plane;
  float*       ob = out + (size_t)b * N3D * plane;

  // Load one (reflect-padded) 18x18 plane tile into buf[sel] (async where available).
  auto load_plane = [&](int sel, int zz) {
    const float* pl = xb + (size_t)zz * plane;
    for (int c = tid; c < LT * LT; c += TS * TS) {
      const int cy = c / LT;
      const int cx = c - cy * LT;
      const int gy = refl(gy0 + cy - 1);
      const int gx = refl(gx0 + cx - 1);
      const float* src = pl + (size_t)gy * N3D + gx;
#if MED3D_ASYNC
      __builtin_amdgcn_global_load_async_to_lds_b32(
          (AS1 int*)(AS1 void*)src,
          (AS3 int*)(AS3 void*)&buf[sel][c], 0, 0);
#else
      buf[sel][c] = *src;
#endif
    }
  };

  auto sync_tile = [&]() {
#if MED3D_ASYNC
    __builtin_amdgcn_s_wait_asynccnt(0);
#endif
    __syncthreads();
  };

  float A[9], Pprev[9], Mm[10];

  // Init: Mm = ranks 4..13 of merge(plane z0-1, plane z0); Pprev = plane z0 (sorted).
  load_plane(0, refl(z0 - 1));
  sync_tile();
  load_plane(1, z0);                 // overlap: issue plane z0 while reading buf0
  sort9_from(buf[0], ty, tx, A);
  sync_tile();
  sort9_from(buf[1], ty, tx, Pprev);
  merge99_mid(A, Pprev, Mm);

  // Prologue of the software pipeline: issue plane z0+1 into buf0.
  load_plane(0, refl(z0 + 1));

  const int zend = z0 + ZCHUNK;      // 160 divisible by ZCHUNK
  for (int z = z0; z < zend; ++z) {
    const int sel = (z - z0) & 1;    // plane z+1 lives in buf[sel]

    // hint the plane after next into the cache hierarchy (global_prefetch_b8)
    {
      const float* pf = xb + (size_t)refl(z + 2) * plane
                           + (size_t)(gy0 + ty) * N3D + (gx0 + tx);
      __builtin_prefetch(pf, 0, 1);
    }

    sync_tile();                     // plane z+1 visible to all waves
    float Pn[9];
    sort9_from(buf[sel], ty, tx, Pn);

    // Overlap: issue async copy of plane z+2 while we crunch the networks.
    if (z + 1 < zend) load_plane(sel ^ 1, refl(z + 2));

    // Exact rank-13 of M(18) U Pn(9), with Mm[i] = M[4+i]:
    // kth smallest = min over i+j=k of max(M[i-1], Pn[j-1]), k = 14.
    float c0 = Mm[9];                 // M[13]
    float c1 = fmaxf(Mm[8], Pn[0]);   // M[12]
    float c2 = fmaxf(Mm[7], Pn[1]);   // M[11]
    float c3 = fmaxf(Mm[6], Pn[2]);   // M[10]
    float c4 = fmaxf(Mm[5], Pn[3]);   // M[9]
    float c5 = fmaxf(Mm[4], Pn[4]);   // M[8]
    float c6 = fmaxf(Mm[3], Pn[5]);   // M[7]
    float c7 = fmaxf(Mm[2], Pn[6]);   // M[6]
    float c8 = fmaxf(Mm[1], Pn[7]);   // M[5]
    float c9 = fmaxf(Mm[0], Pn[8]);   // M[4]
    float m0 = fminf(c0, c1);
    float m1 = fminf(c2, c3);
    float m2 = fminf(c4, c5);
    float m3 = fminf(c6, c7);
    float m4 = fminf(c8, c9);
    float med = fminf(fminf(fminf(m0, m1), fminf(m2, m3)), m4);

    ob[(size_t)z * plane + (size_t)(gy0 + ty) * N3D + (gx0 + tx)] = med;

    if (z + 1 < zend) {
      merge99_mid(Pprev, Pn, Mm);    // Mm <- mid-ranks of merge(plane z, plane z+1)
#pragma unroll
      for (int i = 0; i < 9; ++i) Pprev[i] = Pn[i];
    }
  }
}

extern "C" void kernel_launch(void* const* d_in, const int* in_sizes, int n_in,
                              void* d_out, int out_size, void* d_ws, size_t ws_size,
                              hipStream_t stream) {
  (void)n_in; (void)out_size; (void)d_ws; (void)ws_size;
  const float* x = (const float*)d_in[0];
  float* out = (float*)d_out;
  const int B = in_sizes[0] / (N3D * N3D * N3D);   // = 2
  dim3 block(TS, TS, 1);
  dim3 grid((N3D / TS) * (N3D / TS), N3D / ZCHUNK, B);
  hipLaunchKernelGGL(median_pool3d_68994354642979_kernel, grid, block, 0, stream, x, out);
}